// MambaLayer_76184129896654
// MI455X (gfx1250) — compile-verified
//
#include <hip/hip_runtime.h>
#include <math.h>

typedef __attribute__((ext_vector_type(2))) float v2f;
typedef __attribute__((ext_vector_type(8))) float v8f;

#define B_N   2
#define SEQ   2048
#define MTOT  (B_N*SEQ)      // 4096 rows
#define CDIM  384
#define DIN   768
#define DSTATE 16
#define DTRANK 24
#define DBC_LD 64            // padded x_proj output stride (56 -> 64)

__device__ __forceinline__ float warp_sum(float v) {
  #pragma unroll
  for (int off = 16; off > 0; off >>= 1) v += __shfl_xor(v, off, 32);
  return v;
}

// ---------------- LN1: transpose (B,C,L)->(M,C), layernorm ----------------
__global__ __launch_bounds__(256) void ln1_kernel(const float* __restrict__ x,
    const float* __restrict__ w, const float* __restrict__ b,
    float* __restrict__ xflat, float* __restrict__ xn) {
  int wave = (blockIdx.x * blockDim.x + threadIdx.x) >> 5;
  int lane = threadIdx.x & 31;
  if (wave >= MTOT) return;
  int bb = wave / SEQ, l = wave % SEQ;
  float vals[12];
  float s = 0.f;
  #pragma unroll
  for (int i = 0; i < 12; ++i) {
    int c = lane + 32*i;
    float v = x[((size_t)bb*CDIM + c)*SEQ + l];
    vals[i] = v; s += v;
  }
  s = warp_sum(s);
  float mu = s * (1.f/CDIM);
  float q = 0.f;
  #pragma unroll
  for (int i = 0; i < 12; ++i) { float d = vals[i]-mu; q += d*d; }
  q = warp_sum(q);
  float rs = rsqrtf(q*(1.f/CDIM) + 1e-5f);
  #pragma unroll
  for (int i = 0; i < 12; ++i) {
    int c = lane + 32*i;
    xflat[(size_t)wave*CDIM + c] = vals[i];
    xn[(size_t)wave*CDIM + c]    = (vals[i]-mu)*rs*w[c] + b[c];
  }
}

// ------------- generic fp32 WMMA GEMM: C[M,N] = A[M,K] * W[N,K]^T ----------
// one wave32 computes a 16x64 tile with V_WMMA_F32_16X16X4_F32.
// Software-pipelined: two prefetch stages (k0, k0+4) consumed one
// pair-iteration after issue so loads overlap the matrix pipe.
// unroll(disable) keeps the body at 10 loads + 8 WMMAs so regalloc does not
// recycle load destinations (which would force s_wait_loadcnt 0x0 drains).
// Requires K % 8 == 0 (true for all uses: 384, 768, 24).
__global__ __launch_bounds__(256) void gemm_wmma(const float* __restrict__ A, int lda,
    const float* __restrict__ W, int ldw, float* __restrict__ C, int ldc,
    int M, int N, int K) {
  int wave = (blockIdx.x*blockDim.x + threadIdx.x) >> 5;
  int lane = threadIdx.x & 31;
  int tilesN = N >> 6;
  int tilesM = M >> 4;
  if (wave >= tilesM * tilesN) return;           // wave-uniform guard (EXEC stays full)
  int m0 = (wave / tilesN) << 4;
  int n0 = (wave % tilesN) << 6;
  int lm = lane & 15, half = lane >> 4;

  v8f acc0 = {}, acc1 = {}, acc2 = {}, acc3 = {};
  // A frag: lanes 0-15 -> M rows, K = k0 + {0,1}; lanes 16-31 -> K = k0 + {2,3}
  const float* arow = A + (size_t)(m0 + lm)*lda + 2*half;
  // B frag: lane%16 -> column N, lane/16 selects K pair, VGPR pair = consecutive K
  const float* w0 = W + (size_t)(n0 + lm)*ldw + 2*half;
  const float* w1 = w0 + (size_t)16*ldw;
  const float* w2 = w0 + (size_t)32*ldw;
  const float* w3 = w0 + (size_t)48*ldw;

#define LDA(k) (*(const v2f*)(arow + (k)))
#define LDB0(k) (*(const v2f*)(w0 + (k)))
#define LDB1(k) (*(const v2f*)(w1 + (k)))
#define LDB2(k) (*(const v2f*)(w2 + (k)))
#define LDB3(k) (*(const v2f*)(w3 + (k)))
#define WMMA4(av,b0v,b1v,b2v,b3v)                                                          \
  acc0 = __builtin_amdgcn_wmma_f32_16x16x4_f32(false, av, false, b0v, (short)0, acc0, false, false); \
  acc1 = __builtin_amdgcn_wmma_f32_16x16x4_f32(false, av, false, b1v, (short)0, acc1, false, false); \
  acc2 = __builtin_amdgcn_wmma_f32_16x16x4_f32(false, av, false, b2v, (short)0, acc2, false, false); \
  acc3 = __builtin_amdgcn_wmma_f32_16x16x4_f32(false, av, false, b3v, (short)0, acc3, false, false)

  // prologue: stage A holds k0, stage B holds k0+4
  v2f aA = LDA(0), bA0 = LDB0(0), bA1 = LDB1(0), bA2 = LDB2(0), bA3 = LDB3(0);
  v2f aB = LDA(4), bB0 = LDB0(4), bB1 = LDB1(4), bB2 = LDB2(4), bB3 = LDB3(4);

  int k0 = 0;
  #pragma clang loop unroll(disable)
  for (; k0 < K - 8; k0 += 8) {
    // consume stage A (k = k0), prefetch k0+8 into stage A
    v2f aAn = LDA(k0+8), bA0n = LDB0(k0+8), bA1n = LDB1(k0+8),
        bA2n = LDB2(k0+8), bA3n = LDB3(k0+8);
    WMMA4(aA, bA0, bA1, bA2, bA3);
    aA = aAn; bA0 = bA0n; bA1 = bA1n; bA2 = bA2n; bA3 = bA3n;
    // consume stage B (k = k0+4), prefetch k0+12 into stage B
    v2f aBn = LDA(k0+12), bB0n = LDB0(k0+12), bB1n = LDB1(k0+12),
        bB2n = LDB2(k0+12), bB3n = LDB3(k0+12);
    WMMA4(aB, bB0, bB1, bB2, bB3);
    aB = aBn; bB0 = bB0n; bB1 = bB1n; bB2 = bB2n; bB3 = bB3n;
  }
  // epilogue: k0 == K-8; stages hold K-8 and K-4
  WMMA4(aA, bA0, bA1, bA2, bA3);
  WMMA4(aB, bB0, bB1, bB2, bB3);
#undef LDA
#undef LDB0
#undef LDB1
#undef LDB2
#undef LDB3
#undef WMMA4

  // C/D layout: VGPR v -> M = m0 + v + 8*(lane/16), N = n0 + lane%16
  #pragma unroll
  for (int v = 0; v < 8; ++v) {
    int row = m0 + v + 8*half;
    float* crow = C + (size_t)row*ldc + n0 + lm;
    crow[0]  = acc0[v];
    crow[16] = acc1[v];
    crow[32] = acc2[v];
    crow[48] = acc3[v];
  }
}

// ------------- causal depthwise conv (k=4) + bias + SiLU -------------------
__global__ __launch_bounds__(256) void conv_silu(const float* __restrict__ xz,
    const float* __restrict__ cw, const float* __restrict__ cb,
    float* __restrict__ u) {
  int tid = blockIdx.x*blockDim.x + threadIdx.x;
  if (tid >= MTOT*DIN) return;
  int d = tid % DIN;
  int m = tid / DIN;
  int l = m % SEQ, bb = m / SEQ;
  float acc = cb[d];
  #pragma unroll
  for (int j = 0; j < 4; ++j) {
    int lj = l - 3 + j;
    if (lj >= 0) acc += cw[d*4+j] * xz[(size_t)(bb*SEQ+lj)*(2*DIN) + d];
  }
  u[(size_t)m*DIN + d] = acc / (1.f + __expf(-acc));   // silu
}

// ------------- pad x_proj weight from 56xK to 64xK (zeros) -----------------
__global__ __launch_bounds__(256) void pad_xproj(const float* __restrict__ wsrc,
                                                 float* __restrict__ wdst) {
  int tid = blockIdx.x*blockDim.x + threadIdx.x;
  if (tid >= 64*DIN) return;
  int n = tid / DIN, k = tid % DIN;
  wdst[tid] = (n < (DTRANK + 2*DSTATE)) ? wsrc[(size_t)n*DIN + k] : 0.f;
}

// ------------- delta = softplus(raw + bias) --------------------------------
__global__ __launch_bounds__(256) void softplus_bias(float* __restrict__ delta,
                                                     const float* __restrict__ bias) {
  int tid = blockIdx.x*blockDim.x + threadIdx.x;
  if (tid >= MTOT*DIN) return;
  int d = tid % DIN;
  float xv = delta[tid] + bias[d];
  delta[tid] = fmaxf(xv, 0.f) + log1pf(__expf(-fabsf(xv)));  // stable softplus
}

// ------------- selective scan: one thread per (batch, channel) -------------
__global__ __launch_bounds__(256) void scan_kernel(const float* __restrict__ delta,
    const float* __restrict__ u, const float* __restrict__ dbc,
    const float* __restrict__ xz, const float* __restrict__ A_log,
    const float* __restrict__ Dp, float* __restrict__ yg) {
  int t = blockIdx.x*blockDim.x + threadIdx.x;
  if (t >= B_N*DIN) return;
  int d = t % DIN, bb = t / DIN;
  float Av[DSTATE], h[DSTATE];
  #pragma unroll
  for (int s = 0; s < DSTATE; ++s) { Av[s] = -__expf(A_log[d*DSTATE+s]); h[s] = 0.f; }
  float Dd = Dp[d];
  for (int l = 0; l < SEQ; ++l) {
    size_t m = (size_t)bb*SEQ + l;
    float dl = delta[m*DIN + d];
    float uu = u[m*DIN + d];
    float du = dl * uu;
    const float* bm = dbc + m*DBC_LD + DTRANK;      // B: cols 24..39 (wave-uniform -> broadcast)
    const float* cm = bm + DSTATE;                  // C: cols 40..55
    float y = 0.f;
    #pragma unroll
    for (int s = 0; s < DSTATE; ++s) {
      float dA = __expf(dl * Av[s]);
      h[s] = dA*h[s] + du*bm[s];
      y += h[s]*cm[s];
    }
    float zz = xz[m*(2*DIN) + DIN + d];
    float gate = zz / (1.f + __expf(-zz));
    yg[m*DIN + d] = (y + uu*Dd) * gate;             // fused +u*D and *silu(z)
  }
}

// ------------- residual + LN2 + transpose back to (B,C,L) ------------------
__global__ __launch_bounds__(256) void ln2_out(const float* __restrict__ xflat,
    const float* __restrict__ ymm, const float* __restrict__ w,
    const float* __restrict__ b, float* __restrict__ out) {
  int wave = (blockIdx.x * blockDim.x + threadIdx.x) >> 5;
  int lane = threadIdx.x & 31;
  if (wave >= MTOT) return;
  int bb = wave / SEQ, l = wave % SEQ;
  float vals[12];
  float s = 0.f;
  #pragma unroll
  for (int i = 0; i < 12; ++i) {
    int c = lane + 32*i;
    float v = xflat[(size_t)wave*CDIM + c] + ymm[(size_t)wave*CDIM + c];
    vals[i] = v; s += v;
  }
  s = warp_sum(s);
  float mu = s * (1.f/CDIM);
  float q = 0.f;
  #pragma unroll
  for (int i = 0; i < 12; ++i) { float d = vals[i]-mu; q += d*d; }
  q = warp_sum(q);
  float rs = rsqrtf(q*(1.f/CDIM) + 1e-5f);
  #pragma unroll
  for (int i = 0; i < 12; ++i) {
    int c = lane + 32*i;
    out[((size_t)bb*CDIM + c)*SEQ + l] = (vals[i]-mu)*rs*w[c] + b[c];
  }
}

extern "C" void kernel_launch(void* const* d_in, const int* in_sizes, int n_in,
                              void* d_out, int out_size, void* d_ws, size_t ws_size,
                              hipStream_t stream) {
  const float* x         = (const float*)d_in[0];
  const float* ln1_w     = (const float*)d_in[1];
  const float* ln1_b     = (const float*)d_in[2];
  const float* ln2_w     = (const float*)d_in[3];
  const float* ln2_b     = (const float*)d_in[4];
  const float* in_proj_w = (const float*)d_in[5];   // (1536, 384)
  const float* conv_w    = (const float*)d_in[6];   // (768, 1, 4)
  const float* conv_b    = (const float*)d_in[7];
  const float* x_proj_w  = (const float*)d_in[8];   // (56, 768)
  const float* dt_proj_w = (const float*)d_in[9];   // (768, 24)
  const float* dt_proj_b = (const float*)d_in[10];
  const float* A_log     = (const float*)d_in[11];  // (768, 16)
  const float* Dp        = (const float*)d_in[12];
  const float* out_proj_w= (const float*)d_in[13];  // (384, 768)

  float* ws    = (float*)d_ws;
  float* xflat = ws;                                 // 4096*384
  float* xn    = xflat + (size_t)MTOT*CDIM;          // 4096*384
  float* xz    = xn    + (size_t)MTOT*CDIM;          // 4096*1536
  float* u     = xz    + (size_t)MTOT*2*DIN;         // 4096*768
  float* dbc   = u     + (size_t)MTOT*DIN;           // 4096*64
  float* delta = dbc   + (size_t)MTOT*DBC_LD;        // 4096*768
  float* yg    = delta + (size_t)MTOT*DIN;           // 4096*768
  float* ymm   = yg    + (size_t)MTOT*DIN;           // 4096*384
  float* wpad  = ymm   + (size_t)MTOT*CDIM;          // 64*768

  ln1_kernel<<<MTOT/8, 256, 0, stream>>>(x, ln1_w, ln1_b, xflat, xn);
  pad_xproj<<<(64*DIN+255)/256, 256, 0, stream>>>(x_proj_w, wpad);

  { // xz = xn @ in_proj_w^T : M=4096, N=1536, K=384
    int waves = (MTOT/16)*(2*DIN/64);
    gemm_wmma<<<(waves+7)/8, 256, 0, stream>>>(xn, CDIM, in_proj_w, CDIM, xz, 2*DIN,
                                               MTOT, 2*DIN, CDIM);
  }
  conv_silu<<<(MTOT*DIN+255)/256, 256, 0, stream>>>(xz, conv_w, conv_b, u);

  { // dbc = u @ wpad^T : M=4096, N=64(padded), K=768
    int waves = (MTOT/16)*(64/64);
    gemm_wmma<<<(waves+7)/8, 256, 0, stream>>>(u, DIN, wpad, DIN, dbc, DBC_LD,
                                               MTOT, 64, DIN);
  }
  { // delta_raw = dbc[:, :24] @ dt_proj_w^T : M=4096, N=768, K=24
    int waves = (MTOT/16)*(DIN/64);
    gemm_wmma<<<(waves+7)/8, 256, 0, stream>>>(dbc, DBC_LD, dt_proj_w, DTRANK, delta, DIN,
                                               MTOT, DIN, DTRANK);
  }
  softplus_bias<<<(MTOT*DIN+255)/256, 256, 0, stream>>>(delta, dt_proj_b);
  scan_kernel<<<(B_N*DIN+255)/256, 256, 0, stream>>>(delta, u, dbc, xz, A_log, Dp, yg);

  { // ymm = yg @ out_proj_w^T : M=4096, N=384, K=768
    int waves = (MTOT/16)*(CDIM/64);
    gemm_wmma<<<(waves+7)/8, 256, 0, stream>>>(yg, DIN, out_proj_w, DIN, ymm, CDIM,
                                               MTOT, CDIM, DIN);
  }
  ln2_out<<<MTOT/8, 256, 0, stream>>>(xflat, ymm, ln2_w, ln2_b, (float*)d_out);
}